// SparseDense_12575664242876
// MI455X (gfx1250) — compile-verified
//
#include <hip/hip_runtime.h>
#include <stdint.h>

// out = relu(x @ (W .* mask) + bias)
// x: [16384, 4096] f32, W: [4096, 4096] f32, mask: [4096,4096] bool(u8), bias: [4096] f32
// fp32-exact path via V_WMMA_F32_16X16X4_F32 (gfx1250, wave32).

typedef __attribute__((ext_vector_type(2))) float v2f;
typedef __attribute__((ext_vector_type(8))) float v8f;

#define M_DIM 16384
#define K_DIM 4096
#define N_DIM 4096

#define M_TILE 128      // per block: 8 waves x 16 rows
#define N_TILE 128      // per block (= per wave, split as 8 x 16)
#define K_CHUNK 32      // LDS-staged K slab (8 WMMA steps of K=4)
#define NSUB (N_TILE / 16)

// LDS layout: B slab stored pre-packed for single ds_load_b64 fragments.
//   group g = k4*2 + half   (k4 = k/4, half = (k%4)/2, j = k%2)
//   addr(g, n, j) = g*GROUP_STRIDE + n*2 + j     (float words)
// GROUP_STRIDE = 288 words == 32 (mod 64 banks): lanes 0-15 (half 0) and
// lanes 16-31 (half 1) of one ds_load_b64 hit disjoint bank halves.
#define GROUP_STRIDE 288
#define LDS_WORDS (16 * GROUP_STRIDE)   // 16 groups, 18432 B

__global__ __launch_bounds__(256)
void sparse_dense_wmma_f32(const float* __restrict__ x,
                           const float* __restrict__ w,
                           const unsigned char* __restrict__ mask,
                           const float* __restrict__ bias,
                           float* __restrict__ out)
{
    __shared__ float Bs[LDS_WORDS];

    const int tid  = threadIdx.x;
    const int lane = tid & 31;
    const int wave = tid >> 5;
    const int half = lane >> 4;   // A/B K-pair select: 0 -> K{0,1}, 1 -> K{2,3}
    const int mrow = lane & 15;   // A: M row / B,C,D: N column within 16-wide tile

    const int row0 = blockIdx.y * M_TILE + wave * 16;
    const int col0 = blockIdx.x * N_TILE;

    v8f acc[NSUB] = {};  // eight 16x16 f32 C/D tiles along N

    const float* xrow = x + (size_t)(row0 + mrow) * K_DIM;

    for (int kk = 0; kk < K_DIM; kk += K_CHUNK) {
        // ---- stage masked weight slab (K_CHUNK x N_TILE) into packed LDS ----
        // 4096 f32 elements, 256 threads, 4 contiguous N per thread, 4 reps.
#pragma unroll
        for (int rep = 0; rep < 4; ++rep) {
            const int i = rep * 1024 + tid * 4;       // 0..4095
            const int k = i >> 7;                     // 0..31
            const int n = i & 127;                    // multiple of 4
            const size_t g = (size_t)(kk + k) * N_DIM + (size_t)(col0 + n);
            const float4   wv = *(const float4*)(w + g);      // 16B aligned
            const uint32_t m4 = *(const uint32_t*)(mask + g); // 4B aligned
            const int grp  = ((k >> 2) << 1) + ((k & 3) >> 1);
            const int jbit = k & 1;
            float* dst = Bs + grp * GROUP_STRIDE + n * 2 + jbit;
            dst[0] = (m4 & 0x000000ffu) ? wv.x : 0.0f;
            dst[2] = (m4 & 0x0000ff00u) ? wv.y : 0.0f;
            dst[4] = (m4 & 0x00ff0000u) ? wv.z : 0.0f;
            dst[6] = (m4 & 0xff000000u) ? wv.w : 0.0f;
            // pull next slab of W toward caches while we compute this one
            if (kk + K_CHUNK < K_DIM) {
                __builtin_prefetch(w + g + (size_t)K_CHUNK * N_DIM, 0, 1);
            }
        }
        __syncthreads();

        // ---- 8 WMMA K-steps of 4 over the slab, 8 N-subtiles each ----
#pragma unroll
        for (int k4 = 0; k4 < K_CHUNK / 4; ++k4) {
            // A fragment (16x4 f32): lane half selects K pair, one 8B load
            const v2f a = *(const v2f*)(xrow + kk + k4 * 4 + half * 2);
            const float* bbase = Bs + (k4 * 2 + half) * GROUP_STRIDE;
#pragma unroll
            for (int t = 0; t < NSUB; ++t) {
                // packed pair -> single conflict-free ds_load_b64
                const v2f b = *(const v2f*)(bbase + (t * 16 + mrow) * 2);
                acc[t] = __builtin_amdgcn_wmma_f32_16x16x4_f32(
                    /*neg_a=*/false, a, /*neg_b=*/false, b,
                    /*c_mod=*/(short)0, acc[t],
                    /*reuse_a=*/false, /*reuse_b=*/false);
            }
        }
        __syncthreads();
    }

    // ---- epilogue: bias + ReLU + coalesced store ----
    // C/D layout: VGPR v holds M = row0 + half*8 + v, lane mrow is N column.
#pragma unroll
    for (int t = 0; t < NSUB; ++t) {
        const int n = col0 + t * 16 + mrow;
        const float bv = bias[n];
#pragma unroll
        for (int v = 0; v < 8; ++v) {
            const int m = row0 + half * 8 + v;
            const float r = acc[t][v] + bv;
            out[(size_t)m * N_DIM + n] = r > 0.0f ? r : 0.0f;
        }
    }
}

extern "C" void kernel_launch(void* const* d_in, const int* in_sizes, int n_in,
                              void* d_out, int out_size, void* d_ws, size_t ws_size,
                              hipStream_t stream) {
    const float*         x    = (const float*)d_in[0];
    const float*         w    = (const float*)d_in[1];
    const unsigned char* mask = (const unsigned char*)d_in[2];  // jnp.bool_ = 1 byte
    const float*         bias = (const float*)d_in[3];
    float*               out  = (float*)d_out;

    dim3 grid(N_DIM / N_TILE, M_DIM / M_TILE);  // (32, 128)
    dim3 block(256);
    sparse_dense_wmma_f32<<<grid, block, 0, stream>>>(x, w, mask, bias, out);
}